// Myloss_6408091206114
// MI455X (gfx1250) — compile-verified
//
#include <hip/hip_runtime.h>
#include <cstdint>

#define THREADS     256
#define TILE_CELLS  256
#define CELL_F      30
#define TILE_F      (TILE_CELLS * CELL_F)      // 7680 floats per tensor per tile
#define NCELLS      (2048 * 14 * 14)           // 401408
#define NTILES      (NCELLS / TILE_CELLS)      // 1568
#define BLOCKS      (NTILES / 2)               // 784 -> exactly 2 tiles per block
#define CHUNKS      ((TILE_F / 2) / THREADS)   // 15 b64 chunks per thread per tensor

typedef __attribute__((ext_vector_type(2))) float v2f;
typedef __attribute__((ext_vector_type(8))) float v8f;
typedef __attribute__((ext_vector_type(2))) int   v2i;

#if defined(__gfx1250__) && __has_builtin(__builtin_amdgcn_global_load_async_to_lds_b64)
#define HAVE_ASYNC 1
typedef __attribute__((address_space(1))) v2i g_v2i;   // global (AS1) 2 x i32
typedef __attribute__((address_space(3))) v2i l_v2i;   // LDS (AS3) 2 x i32
static __device__ __forceinline__ g_v2i* to_glb(const void* p) {
  return (g_v2i*)(unsigned long long)(uintptr_t)p;     // generic global addr == AS1 addr
}
static __device__ __forceinline__ l_v2i* to_lds(const void* p) {
  return (l_v2i*)(unsigned)(uintptr_t)p;               // low 32 bits of generic LDS ptr == LDS offset
}
static __device__ __forceinline__ void wait_async0() {
#if __has_builtin(__builtin_amdgcn_s_wait_asynccnt)
  __builtin_amdgcn_s_wait_asynccnt(0);
#else
  asm volatile("s_wait_asynccnt 0" ::: "memory");
#endif
}
#endif

__global__ __launch_bounds__(THREADS) void yolo_loss_kernel(
    const float* __restrict__ pred, const float* __restrict__ targ,
    float* __restrict__ out) {
  __shared__ float lp[TILE_F];
  __shared__ float lt[TILE_F];
  __shared__ float bsum;

  const int tid = threadIdx.x;
  if (tid == 0) bsum = 0.0f;

  float acc = 0.0f;

  for (int tile = blockIdx.x; tile < NTILES; tile += gridDim.x) {
    const float* gp = pred + (size_t)tile * TILE_F;
    const float* gt = targ + (size_t)tile * TILE_F;

    __syncthreads();  // previous tile fully consumed before overwrite
#ifdef HAVE_ASYNC
#pragma unroll
    for (int i = 0; i < CHUNKS; ++i) {
      const int c = 2 * (tid + i * THREADS);  // float index, 8B aligned
      __builtin_amdgcn_global_load_async_to_lds_b64(to_glb(gp + c), to_lds(&lp[c]), 0, 0);
      __builtin_amdgcn_global_load_async_to_lds_b64(to_glb(gt + c), to_lds(&lt[c]), 0, 0);
    }
    wait_async0();  // each wave waits its own ASYNCcnt, barrier joins waves
#else
#pragma unroll
    for (int i = 0; i < CHUNKS; ++i) {
      const int c = 2 * (tid + i * THREADS);
      *(float2*)&lp[c] = *(const float2*)(gp + c);
      *(float2*)&lt[c] = *(const float2*)(gt + c);
    }
#endif
    __syncthreads();

    const int o = tid * CELL_F;

    // target box 0 corners (xyxy) and area
    const float tx = lt[o + 0], ty = lt[o + 1], tw = lt[o + 2], th = lt[o + 3];
    const float tltx = tx - 0.5f * tw, tlty = ty - 0.5f * th;
    const float trbx = tx + 0.5f * tw, trby = ty + 0.5f * th;
    const float area2 = tw * th;

    float iou[2], loc[2], con[2];
#pragma unroll
    for (int b = 0; b < 2; ++b) {
      const int q = o + 5 * b;
      const float px = lp[q + 0], py = lp[q + 1], pw = lp[q + 2], ph = lp[q + 3];
      const float ltx = fmaxf(px - 0.5f * pw, tltx);
      const float lty = fmaxf(py - 0.5f * ph, tlty);
      const float rbx = fminf(px + 0.5f * pw, trbx);
      const float rby = fminf(py + 0.5f * ph, trby);
      // faithful to the reference bug: indicator of (rb - lt < 0)
      const float whx = (rbx - ltx < 0.0f) ? 1.0f : 0.0f;
      const float why = (rby - lty < 0.0f) ? 1.0f : 0.0f;
      const float inter = whx * why;
      const float area1 = pw * ph;
      iou[b] = inter / (area1 + area2 - inter);

      const float dx = px - lt[q + 0];
      const float dy = py - lt[q + 1];
      const float dw = sqrtf(pw) - sqrtf(lt[q + 2]);
      const float dh = sqrtf(ph) - sqrtf(lt[q + 3]);
      loc[b] = dx * dx + dy * dy + dw * dw + dh * dh;
      const float dc = lp[q + 4] - lt[q + 4];
      con[b] = dc * dc;
    }

    // argmax tie-break: box 0 wins on equality (jnp.argmax -> first index)
    const float r0 = (iou[0] >= iou[1]) ? 1.0f : 0.0f;
    const float r1 = 1.0f - r0;

    float cls = 0.0f;
#pragma unroll
    for (int k = 10; k < 30; ++k) {
      const float d = lp[o + k] - lt[o + k];
      cls = fmaf(d, d, cls);
    }

    const float t4 = lt[o + 4];
    const float coo = (t4 > 0.0f) ? 1.0f : 0.0f;
    const float noo = (t4 == 0.0f) ? 1.0f : 0.0f;
    const float d4 = lp[o + 4] - lt[o + 4];
    const float d9 = lp[o + 9] - lt[o + 9];

    // loc + contain + not_contain(==contain) + nooobj + class
    acc += coo * (r0 * (loc[0] + 2.0f * con[0]) + r1 * (loc[1] + 2.0f * con[1]) + cls) +
           noo * (d4 * d4 + d9 * d9);
  }

  const float part = acc * (1.0f / 2048.0f);

#if defined(__gfx1250__) && __has_builtin(__builtin_amdgcn_wmma_f32_16x16x4_f32)
  // Wave reduction with the f32 matrix engine (full precision, RNE):
  // A[m][0] = partial of lane m (lanes 0-15), A[m][2] = partial of lane m+16; B = ones
  // -> D[m][n] = partial_m + partial_{m+16} for every n.
  v2f a = {part, 0.0f};
  v2f bb = {1.0f, 1.0f};
  v8f c = {};
  c = __builtin_amdgcn_wmma_f32_16x16x4_f32(false, a, false, bb, (short)0, c, false, false);
  // lane 0 holds D[0..7][0], lane 16 holds D[8..15][0]; their sums cover the wave
  const float s8 = ((c[0] + c[1]) + (c[2] + c[3])) + ((c[4] + c[5]) + (c[6] + c[7]));
  const int lane = tid & 31;
  if (lane == 0 || lane == 16) atomicAdd(&bsum, s8);
#else
  float v = part;
  for (int off = 16; off > 0; off >>= 1) v += __shfl_xor(v, off, 32);
  if ((tid & 31) == 0) atomicAdd(&bsum, v);
#endif

  __syncthreads();
  if (tid == 0) atomicAdd(out, bsum);  // one global atomic per block (784 total)
}

__global__ void zero_out_kernel(float* out) { out[0] = 0.0f; }

extern "C" void kernel_launch(void* const* d_in, const int* in_sizes, int n_in,
                              void* d_out, int out_size, void* d_ws, size_t ws_size,
                              hipStream_t stream) {
  (void)in_sizes; (void)n_in; (void)out_size; (void)d_ws; (void)ws_size;
  const float* pred = (const float*)d_in[0];
  const float* targ = (const float*)d_in[1];
  float* out = (float*)d_out;

  zero_out_kernel<<<1, 1, 0, stream>>>(out);
  yolo_loss_kernel<<<BLOCKS, THREADS, 0, stream>>>(pred, targ, out);
}